// ContrastiveCrossEntropyLoss_2594160247130
// MI455X (gfx1250) — compile-verified
//
#include <hip/hip_runtime.h>

// ContrastiveCrossEntropyLoss, N=8192 rows, V=32000 classes, k=2.
// Algebraic identity: per-row loss = top1(logits) + top2(logits) - 2*logits[target]
// (the log-sum-exp terms cancel exactly), so this is a pure HBM streaming
// reduction: 1.05 GB @ 23.3 TB/s => ~45 us floor. We stream each row through
// LDS with CDNA5 async global->LDS DMA (ASYNCcnt), double-buffered per wave,
// with non-temporal hints since the stream is single-use (5x the 192MB L2).

#define VCLS 32000
#define NROWS 8192
#define WAVES_PER_BLOCK 8
#define CHUNK_BYTES 512                     // 32 lanes x 16B per async b128
#define CHUNKS_PER_STAGE 5
#define STAGE_BYTES (CHUNKS_PER_STAGE * CHUNK_BYTES)   // 2560 B = 640 floats
#define STAGE_FLOATS (STAGE_BYTES / 4)                 // 640
#define NSTAGES (VCLS / STAGE_FLOATS)                  // 50 (exact)

typedef __attribute__((address_space(3))) char lds_char;

__device__ __forceinline__ unsigned lds_offset(void* p) {
    // generic -> LDS addrspace cast; AS(3) pointers are 32-bit LDS byte offsets
    return (unsigned)(unsigned long long)(lds_char*)p;
}

// Async DMA of 16B per lane: copies from global base saddr plus offset voff to LDS address ldsaddr. Tracked by
// ASYNCcnt. Non-temporal: single-use stream, don't hold it in cache.
#define ASYNC_B128(ldsaddr, voff, saddr)                                          \
    asm volatile("global_load_async_to_lds_b128 %0, %1, %2 th:TH_LOAD_NT"         \
                 :: "v"(ldsaddr), "v"(voff), "s"(saddr) : "memory")

__global__ __launch_bounds__(256) void cce_rows_kernel(const float* __restrict__ logits,
                                                       const int* __restrict__ target,
                                                       float* __restrict__ row_val) {
    __shared__ __align__(16) float smem[WAVES_PER_BLOCK][2][STAGE_FLOATS];

    const int lane = threadIdx.x & 31;
    const int wave = threadIdx.x >> 5;
    const int row  = blockIdx.x * WAVES_PER_BLOCK + wave;
    const float* rowp = logits + (size_t)row * VCLS;   // wave-uniform -> SGPR pair

    const unsigned lane16   = (unsigned)lane * 16u;
    const unsigned lds_base = lds_offset(&smem[wave][0][0]);

    float m1 = -__builtin_inff();
    float m2 = -__builtin_inff();

    // Prologue: DMA stage 0 into buffer 0.
#pragma unroll
    for (int c = 0; c < CHUNKS_PER_STAGE; ++c) {
        ASYNC_B128(lds_base + (unsigned)(c * CHUNK_BYTES) + lane16,
                   (unsigned)(c * CHUNK_BYTES) + lane16, rowp);
    }

    for (int s = 0; s < NSTAGES; ++s) {
        const int buf = s & 1;
        if (s + 1 < NSTAGES) {
            // Prefetch next stage into the other buffer while we reduce this one.
            const unsigned gbase = (unsigned)((s + 1) * STAGE_BYTES);
            const unsigned lbase = lds_base + (unsigned)(((s + 1) & 1) * STAGE_BYTES);
#pragma unroll
            for (int c = 0; c < CHUNKS_PER_STAGE; ++c) {
                ASYNC_B128(lbase + (unsigned)(c * CHUNK_BYTES) + lane16,
                           gbase + (unsigned)(c * CHUNK_BYTES) + lane16, rowp);
            }
            // Async loads retire in order: <=5 outstanding => current stage landed.
            asm volatile("s_wait_asynccnt 0x5" ::: "memory");
        } else {
            asm volatile("s_wait_asynccnt 0x0" ::: "memory");
        }

        const float4* bp = (const float4*)&smem[wave][buf][0];
#pragma unroll
        for (int c = 0; c < CHUNKS_PER_STAGE; ++c) {
            float4 v = bp[c * 32 + lane];          // ds_load_b128, conflict-free
            m2 = fmaxf(m2, fminf(m1, v.x)); m1 = fmaxf(m1, v.x);
            m2 = fmaxf(m2, fminf(m1, v.y)); m1 = fmaxf(m1, v.y);
            m2 = fmaxf(m2, fminf(m1, v.z)); m1 = fmaxf(m1, v.z);
            m2 = fmaxf(m2, fminf(m1, v.w)); m1 = fmaxf(m1, v.w);
        }
    }

    // Wave32 top-2 merge: top2(A ∪ B) from per-lane (m1,m2) pairs.
#pragma unroll
    for (int off = 16; off >= 1; off >>= 1) {
        float o1 = __shfl_xor(m1, off, 32);
        float o2 = __shfl_xor(m2, off, 32);
        float n1 = fmaxf(m1, o1);
        float n2 = fmaxf(fminf(m1, o1), fmaxf(m2, o2));
        m1 = n1; m2 = n2;
    }

    if (lane == 0) {
        float xt = rowp[target[row]];
        row_val[row] = m1 + m2 - 2.0f * xt;
    }
}

// Deterministic final reduction of 8192 per-row values (no float atomics).
__global__ __launch_bounds__(256) void cce_reduce_kernel(const float* __restrict__ row_val,
                                                         float* __restrict__ out) {
    __shared__ float sm[256];
    float s = 0.0f;
    for (int i = threadIdx.x; i < NROWS; i += 256) s += row_val[i];
    sm[threadIdx.x] = s;
    __syncthreads();
#pragma unroll
    for (int w = 128; w >= 1; w >>= 1) {
        if ((int)threadIdx.x < w) sm[threadIdx.x] += sm[threadIdx.x + w];
        __syncthreads();
    }
    if (threadIdx.x == 0) out[0] = sm[0];
}

extern "C" void kernel_launch(void* const* d_in, const int* in_sizes, int n_in,
                              void* d_out, int out_size, void* d_ws, size_t ws_size,
                              hipStream_t stream) {
    const float* logits = (const float*)d_in[0];
    const int*   target = (const int*)d_in[1];
    float* rows = (float*)d_ws;                       // 8192 floats of scratch

    cce_rows_kernel<<<NROWS / WAVES_PER_BLOCK, 256, 0, stream>>>(logits, target, rows);
    cce_reduce_kernel<<<1, 256, 0, stream>>>(rows, (float*)d_out);
}